// SGL_Encoder_83949430767919
// MI455X (gfx1250) — compile-verified
//
#include <hip/hip_runtime.h>
#include <hip/hip_bf16.h>

#ifndef __has_builtin
#define __has_builtin(x) 0
#endif

#if __has_builtin(__builtin_amdgcn_global_load_async_to_lds_b32) && \
    __has_builtin(__builtin_amdgcn_s_wait_asynccnt)
#define USE_ASYNC_LDS 1
#else
#define USE_ASYNC_LDS 0
#endif

#define NUSERS 100000
#define NITEMS 200000
#define NNODES (NUSERS + NITEMS)
#define EMB    64
#define CHUNK  1024
#define BLOCK  256
#define OPS_PER_CHUNK ((CHUNK / BLOCK) * 3)   // 12 async b32 issues per thread

#if USE_ASYNC_LDS
typedef __attribute__((address_space(1))) int* gas1_i32_p;
typedef __attribute__((address_space(3))) int* as3_i32_p;

// Async global->LDS copy of one b32 per lane (ASYNCcnt-tracked, no VGPR data).
__device__ __forceinline__ void async_copy_b32(const void* gsrc, void* ldst) {
    __builtin_amdgcn_global_load_async_to_lds_b32(
        (gas1_i32_p)(int*)gsrc, (as3_i32_p)(int*)ldst, 0, 0);
}
#endif

// ---------------------------------------------------------------------------
// COO SpMM: y[row] += val * x[col].  One wave per edge; 32 lanes cover the
// 64-float embedding row as float2 each (256B coalesced gather + scatter).
// Edge metadata is streamed into LDS in 1024-edge chunks through the CDNA5
// async global->LDS path (ASYNCcnt), double-buffered: chunk k+1 streams in
// while chunk k is processed; s_wait_asynccnt(12) drains only the older ops.
// Per-thread issue count is kept constant (indices clamped, not predicated)
// so the wait immediate is exact; over-staged entries are never read.
// ---------------------------------------------------------------------------
__global__ __launch_bounds__(BLOCK) void spmm_coo_kernel(
    const int*   __restrict__ rows,
    const int*   __restrict__ cols,
    const float* __restrict__ vals,
    const float* __restrict__ x,
    float*       __restrict__ y,
    int nnz)
{
    __shared__ int   s_row[2][CHUNK];
    __shared__ int   s_col[2][CHUNK];
    __shared__ float s_val[2][CHUNK];

    const int tid  = threadIdx.x;
    const int wave = tid >> 5;
    const int lane = tid & 31;

    int base = (int)blockIdx.x * CHUNK;
    const int stride = (int)gridDim.x * CHUNK;
    if (base >= nnz) return;

    // Stage a chunk into buffer b: always 12 async ops per thread.
    auto stage = [&](int b, int cbase) {
#if USE_ASYNC_LDS
        #pragma unroll
        for (int k = 0; k < CHUNK / BLOCK; ++k) {
            const int i = tid + k * BLOCK;
            int g = cbase + i;
            g = (g < nnz) ? g : (nnz - 1);          // clamp, keep op count fixed
            async_copy_b32(rows + g, &s_row[b][i]);
            async_copy_b32(cols + g, &s_col[b][i]);
            async_copy_b32(vals + g, &s_val[b][i]);
        }
#else
        #pragma unroll
        for (int k = 0; k < CHUNK / BLOCK; ++k) {
            const int i = tid + k * BLOCK;
            int g = cbase + i;
            g = (g < nnz) ? g : (nnz - 1);
            s_row[b][i] = rows[g];
            s_col[b][i] = cols[g];
            s_val[b][i] = vals[g];
        }
#endif
    };

    stage(0, base);                                  // prologue
    int b = 0;
    for (; base < nnz; base += stride, b ^= 1) {
        const int nxt = base + stride;
        const bool have_next = (nxt < nnz);
        if (have_next) stage(b ^ 1, nxt);            // issue-ahead
#if USE_ASYNC_LDS
        if (have_next) __builtin_amdgcn_s_wait_asynccnt(OPS_PER_CHUNK);
        else           __builtin_amdgcn_s_wait_asynccnt(0);
#endif
        __syncthreads();

        const int cnt = min(CHUNK, nnz - base);
        const int e0  = wave * (CHUNK / 8);
        const int e1  = min(e0 + (CHUNK / 8), cnt);
        #pragma unroll 4
        for (int e = e0; e < e1; ++e) {
            if (e + 8 < e1) {
                // global_prefetch_b8 on an upcoming gather row
                __builtin_prefetch(
                    &x[(size_t)s_col[b][e + 8] * EMB + 2 * lane], 0, 0);
            }
            const int   r = s_row[b][e];
            const int   c = s_col[b][e];
            const float v = s_val[b][e];

            const float2 xv = *reinterpret_cast<const float2*>(
                x + (size_t)c * EMB + 2 * lane);
            float* yp = y + (size_t)r * EMB + 2 * lane;
            unsafeAtomicAdd(yp,     v * xv.x);       // hw global_atomic_add_f32
            unsafeAtomicAdd(yp + 1, v * xv.y);
        }
        __syncthreads();                             // buffer b free for restage
    }
}

// ---------------------------------------------------------------------------
// Fused init: xa = acc = concat(user_emb, item_emb); xb = 0.   float4 I/O.
// ---------------------------------------------------------------------------
__global__ __launch_bounds__(BLOCK) void init_kernel(
    const float4* __restrict__ ue, const float4* __restrict__ ie,
    float4* __restrict__ xa, float4* __restrict__ acc,
    float4* __restrict__ xb, int nu4, int total4)
{
    for (int i = blockIdx.x * BLOCK + threadIdx.x; i < total4;
         i += gridDim.x * BLOCK) {
        float4 v = (i < nu4) ? ue[i] : ie[i - nu4];
        xa[i]  = v;
        acc[i] = v;
        xb[i]  = make_float4(0.f, 0.f, 0.f, 0.f);
    }
}

// ---------------------------------------------------------------------------
// Fused: acc += s;  z = 0  (zeroes the buffer the next SpMM scatters into).
// ---------------------------------------------------------------------------
__global__ __launch_bounds__(BLOCK) void add_zero_kernel(
    float4* __restrict__ acc, const float4* __restrict__ s,
    float4* __restrict__ z, int total4)
{
    for (int i = blockIdx.x * BLOCK + threadIdx.x; i < total4;
         i += gridDim.x * BLOCK) {
        float4 a = acc[i];
        float4 bb = s[i];
        a.x += bb.x; a.y += bb.y; a.z += bb.z; a.w += bb.w;
        acc[i] = a;
        z[i]   = make_float4(0.f, 0.f, 0.f, 0.f);
    }
}

// ---------------------------------------------------------------------------
// Fused final layer accumulate + scale:  out = (acc + s) * 0.25
// d_out = user_all ++ item_all  ==  the full all_emb array, flat.
// ---------------------------------------------------------------------------
__global__ __launch_bounds__(BLOCK) void final_kernel(
    const float4* __restrict__ acc, const float4* __restrict__ s,
    float4* __restrict__ out, int total4)
{
    for (int i = blockIdx.x * BLOCK + threadIdx.x; i < total4;
         i += gridDim.x * BLOCK) {
        float4 a = acc[i];
        float4 bb = s[i];
        a.x = (a.x + bb.x) * 0.25f;
        a.y = (a.y + bb.y) * 0.25f;
        a.z = (a.z + bb.z) * 0.25f;
        a.w = (a.w + bb.w) * 0.25f;
        out[i] = a;
    }
}

extern "C" void kernel_launch(void* const* d_in, const int* in_sizes, int n_in,
                              void* d_out, int out_size, void* d_ws, size_t ws_size,
                              hipStream_t stream)
{
    const float* user_emb = (const float*)d_in[0];
    const float* item_emb = (const float*)d_in[1];
    const float* adj_vals = (const float*)d_in[2];
    const int*   adj_row  = (const int*)d_in[3];
    const int*   adj_col  = (const int*)d_in[4];
    const int    nnz      = in_sizes[2];

    // Workspace layout: xa | xb | acc  (3 x 300000*64 f32 = 230.4 MB)
    const size_t nflt = (size_t)NNODES * EMB;
    float* xa  = (float*)d_ws;
    float* xb  = xa + nflt;
    float* acc = xb + nflt;

    const int total4 = (int)(nflt / 4);
    const int nu4    = NUSERS * EMB / 4;

    const int gElem  = (total4 + BLOCK - 1) / BLOCK;
    const int chunks = (nnz + CHUNK - 1) / CHUNK;
    const int gSpmm  = chunks < 1024 ? chunks : 1024;   // ~2 chunks/block -> pipeline

    init_kernel<<<gElem, BLOCK, 0, stream>>>(
        (const float4*)user_emb, (const float4*)item_emb,
        (float4*)xa, (float4*)acc, (float4*)xb, nu4, total4);

    // layer 1: xa -> xb ; acc += xb ; zero xa
    spmm_coo_kernel<<<gSpmm, BLOCK, 0, stream>>>(adj_row, adj_col, adj_vals,
                                                 xa, xb, nnz);
    add_zero_kernel<<<gElem, BLOCK, 0, stream>>>((float4*)acc, (const float4*)xb,
                                                 (float4*)xa, total4);
    // layer 2: xb -> xa ; acc += xa ; zero xb
    spmm_coo_kernel<<<gSpmm, BLOCK, 0, stream>>>(adj_row, adj_col, adj_vals,
                                                 xb, xa, nnz);
    add_zero_kernel<<<gElem, BLOCK, 0, stream>>>((float4*)acc, (const float4*)xa,
                                                 (float4*)xb, total4);
    // layer 3: xa -> xb ; out = (acc + xb) / 4
    spmm_coo_kernel<<<gSpmm, BLOCK, 0, stream>>>(adj_row, adj_col, adj_vals,
                                                 xa, xb, nnz);
    final_kernel<<<gElem, BLOCK, 0, stream>>>((const float4*)acc,
                                              (const float4*)xb,
                                              (float4*)d_out, total4);
}